// TokenClusterInter_72361609003071
// MI455X (gfx1250) — compile-verified
//
#include <hip/hip_runtime.h>

typedef float v2f __attribute__((ext_vector_type(2)));
typedef float v8f __attribute__((ext_vector_type(8)));

#define NPTS   147          // points per segment (FD * 49)
#define NPAD   160          // padded to 10 WMMA tiles
#define NCLU   49           // medoids
#define KPAD   64           // padded to 4 WMMA tiles
#define WIDTH  768          // D
#define BT     768          // B * T_BEFORE (64*12)
#define DCHUNK 64           // D staged per LDS chunk
#define NCHUNK (WIDTH / DCHUNK)   // 12
#define XPITCH 68           // Xs row pitch (floats) -> conflict-free, 272B = 16B-aligned rows
#define DPITCH 148          // dist row pitch (floats) -> conflict-free
#define OPITCH 164          // one-hot^T row pitch -> conflict-free
#define ITERS  15

// ---- LDS layout (float offsets) ----
#define OFF_DIST  0                         // 23680 floats
#define OFF_XS0   (NPAD * DPITCH)           // 23680 ; chunk buffer 0 (10880) / ohT overlap
#define OFF_XS1   (OFF_XS0 + NPAD * XPITCH) // 34560 ; chunk buffer 1 (10880)
#define OFF_SQ    (OFF_XS1 + NPAD * XPITCH) // 45440
#define OFF_MIND  (OFF_SQ + 160)            // 45600
#define OFF_KEYS  (OFF_MIND + 160)          // 45760 (8B aligned)
#define OFF_MED   (OFF_KEYS + 128)          // 45888 (u64[64] = 128 floats)
#define OFF_ROW   (OFF_MED + 64)            // 45952
#define OFF_RED   (OFF_ROW + 160)           // 46112
#define LDS_FLOATS (OFF_RED + 8)            // 46120 -> 184480 bytes

#define WAIT_ASYNC(N) asm volatile("s_wait_asynccnt " #N ::: "memory")

__device__ __forceinline__ v8f v8f_zero() {
  v8f z;
#pragma unroll
  for (int i = 0; i < 8; ++i) z[i] = 0.f;
  return z;
}

// argmax over arr[0..146], first-index tie-break (wave-0 only, wave32)
__device__ __forceinline__ int argmax147(const float* arr, int lane) {
  float bv = -3.4e38f;
  int bidx = 0x7fffffff;
  for (int n = lane; n < NPTS; n += 32) {
    float v = arr[n];
    if (v > bv) { bv = v; bidx = n; }   // per-lane n ascending -> first max kept
  }
#pragma unroll
  for (int off = 16; off > 0; off >>= 1) {
    float ov = __shfl_xor(bv, off);
    int   oi = __shfl_xor(bidx, off);
    if (ov > bv || (ov == bv && oi < bidx)) { bv = ov; bidx = oi; }
  }
  return bidx;
}

// Issue one chunk's async global->LDS stage: 2560 B128 ops (10 per thread, uniform
// per wave so ASYNCcnt bookkeeping is exact). Rows >= NPTS pull from a safe
// address (rowoff=0); their garbage lands only in pad rows whose outputs are
// discarded.
__device__ __forceinline__ void async_stage(const float* __restrict__ x,
                                            const int* rowoff, float* XsBuf,
                                            int dc, int tid) {
#pragma unroll
  for (int i = 0; i < 10; ++i) {
    int idx = tid + 256 * i;
    int row = idx >> 4;                 // 16 float4 per row
    int c4  = idx & 15;
    const float* g = x + rowoff[row] + dc * DCHUNK + c4 * 4;
    unsigned ldsaddr = (unsigned)(unsigned long long)(XsBuf + row * XPITCH + c4 * 4);
    asm volatile("global_load_async_to_lds_b128 %0, %1, off"
                 :: "v"(ldsaddr), "v"(g) : "memory");
  }
}

__global__ __launch_bounds__(256)
void tci_kernel(const float* __restrict__ x, float* __restrict__ out) {
  extern __shared__ float lds[];
  float* distM = lds + OFF_DIST;
  float* Xs0   = lds + OFF_XS0;         // phase A double buffer
  float* Xs1   = lds + OFF_XS1;
  float* ohT   = lds + OFF_XS0;         // phase D one-hot^T (reuses Xs0 space)
  float* sq    = lds + OFF_SQ;
  float* mind  = lds + OFF_MIND;
  unsigned long long* keys = (unsigned long long*)(lds + OFF_KEYS);
  int* med    = (int*)(lds + OFF_MED);
  int* rowoff = (int*)(lds + OFF_ROW);
  int* red    = (int*)(lds + OFF_RED);

  const int tid  = threadIdx.x;
  const int wave = tid >> 5;
  const int lane = tid & 31;
  const int lr   = lane & 15;
  const int hi   = lane >> 4;

  const int m      = blockIdx.x;        // segment id = ta*64 + bi
  const int bi     = m & 63;
  const int ta     = m >> 6;
  const int frame0 = bi * 12 + ta * 3;  // first frame of this temporal segment

  // per-point global row offsets (floats): point n -> x[p+1, frame0+fd, :]
  if (tid < NPAD) {
    int off = 0;
    if (tid < NPTS) {
      int fd = tid / 49;
      int p  = tid - fd * 49;
      off = ((p + 1) * BT + (frame0 + fd)) * WIDTH;
    }
    rowoff[tid] = off;
  }

  // this wave's upper-triangle tiles: u = wave + 8*s (u < 55)
  int tiA[7], tjA[7];
#pragma unroll
  for (int s = 0; s < 7; ++s) {
    int uu = wave + 8 * s, ti = 0, rem = 10;
    while (uu >= rem) { uu -= rem; rem--; ti++; }
    tiA[s] = ti; tjA[s] = ti + uu;
  }

  v8f acc[7];
#pragma unroll
  for (int s = 0; s < 7; ++s) acc[s] = v8f_zero();

  __syncthreads();                      // rowoff visible to all waves

  // ====== Phase A: Gram G = X X^T ; async double-buffered staging ======
  async_stage(x, rowoff, Xs0, 0, tid);  // prefetch chunk 0
#pragma unroll 1
  for (int dc = 0; dc < NCHUNK; ++dc) {
    float* cur = (dc & 1) ? Xs1 : Xs0;
    if (dc + 1 < NCHUNK) {
      // previous iteration's trailing barrier guarantees nobody still reads nxt
      float* nxt = (dc & 1) ? Xs0 : Xs1;
      async_stage(x, rowoff, nxt, dc + 1, tid);
      WAIT_ASYNC(10);                   // in-order: chunk dc's 10 ops are done
    } else {
      WAIT_ASYNC(0);
    }
    __syncthreads();                    // every wave waited -> chunk dc landed

#pragma unroll
    for (int s = 0; s < 7; ++s) {
      if (wave + 8 * s < 55) {
        const float* arow = cur + (tiA[s] * 16 + lr) * XPITCH + 2 * hi;
        const float* brow = cur + (tjA[s] * 16 + lr) * XPITCH + 2 * hi;
        v8f c = acc[s];
#pragma unroll
        for (int kk = 0; kk < DCHUNK; kk += 4) {
          v2f a = *(const v2f*)(arow + kk);
          v2f b = *(const v2f*)(brow + kk);
          c = __builtin_amdgcn_wmma_f32_16x16x4_f32(false, a, false, b,
                                                    (short)0, c, false, false);
        }
        acc[s] = c;
      }
    }
    __syncthreads();                    // protect cur before it is re-issued
  }

  // ================= Phase B: distances into LDS =================
  {
    float4 z4; z4.x = z4.y = z4.z = z4.w = 0.f;
    for (int i = tid; i < NPAD * DPITCH / 4; i += 256)
      ((float4*)distM)[i] = z4;         // pads must be 0 for the intra WMMA
  }
  __syncthreads();
#pragma unroll
  for (int s = 0; s < 7; ++s) {                 // squared norms = Gram diagonal
    if (wave + 8 * s < 55 && tiA[s] == tjA[s]) {
#pragma unroll
      for (int v = 0; v < 8; ++v)
        if (lr == v + 8 * hi) sq[tiA[s] * 16 + v + 8 * hi] = acc[s][v];
    }
  }
  __syncthreads();
#pragma unroll
  for (int s = 0; s < 7; ++s) {
    if (wave + 8 * s < 55) {
#pragma unroll
      for (int v = 0; v < 8; ++v) {
        int gi = tiA[s] * 16 + v + 8 * hi;
        int gj = tjA[s] * 16 + lr;
        if (gi < NPTS && gj < NPTS) {
          float d = sqrtf(fmaxf(sq[gi] + sq[gj] - 2.f * acc[s][v], 0.f));
          distM[gi * DPITCH + gj] = d;
          distM[gj * DPITCH + gi] = d;
        }
      }
    }
  }
  __syncthreads();

  // ================= Phase C: KKZ init =================
  if (wave == 0) {
    int first = argmax147(sq, lane);            // argmax ||x|| == argmax ||x||^2
    if (lane == 0) { med[0] = first; red[0] = first; }
  }
  __syncthreads();
  if (tid < NPTS) mind[tid] = distM[tid * DPITCH + red[0]];
  __syncthreads();
#pragma unroll 1
  for (int i = 1; i < NCLU; ++i) {
    if (wave == 0) {
      int nxt = argmax147(mind, lane);
      if (lane == 0) { med[i] = nxt; red[0] = nxt; }
    }
    __syncthreads();
    int nxt = red[0];
    if (tid < NPTS) mind[tid] = fminf(mind[tid], distM[tid * DPITCH + nxt]);
    __syncthreads();
  }

  // ================= Phase D: 15 k-medoids iterations =================
#pragma unroll 1
  for (int it = 0; it < ITERS; ++it) {
    {
      float4 z4; z4.x = z4.y = z4.z = z4.w = 0.f;
      for (int i = tid; i < KPAD * OPITCH / 4; i += 256)
        ((float4*)ohT)[i] = z4;
    }
    if (tid < KPAD)                                        // empty cluster -> (1e10, n=0)
      keys[tid] = ((unsigned long long)__float_as_uint(1e10f)) << 32;
    __syncthreads();

    if (tid < NPTS) {                                      // assignment (first-min)
      float best = 3.4e38f; int bk = 0;
      for (int k = 0; k < NCLU; ++k) {
        float dv = distM[tid * DPITCH + med[k]];
        if (dv < best) { best = dv; bk = k; }
      }
      ohT[bk * OPITCH + tid] = 1.f;
    }
    __syncthreads();

    // intra = dist(160x160) x onehot(160x64) via WMMA (pads are zero)
    v8f ic[5];
#pragma unroll
    for (int s = 0; s < 5; ++s) {
      int t  = wave + 8 * s;                               // exactly 40 tiles / 8 waves
      int ri = t >> 2, cj = t & 3;
      const float* arow = distM + (ri * 16 + lr) * DPITCH + 2 * hi;
      const float* brow = ohT   + (cj * 16 + lr) * OPITCH + 2 * hi;
      v8f c = v8f_zero();
#pragma unroll 4
      for (int mm = 0; mm < NPAD; mm += 4) {
        v2f a = *(const v2f*)(arow + mm);
        v2f b = *(const v2f*)(brow + mm);
        c = __builtin_amdgcn_wmma_f32_16x16x4_f32(false, a, false, b,
                                                  (short)0, c, false, false);
      }
      ic[s] = c;
    }

    // member-masked argmin over n per cluster via packed u64 LDS atomic-min
#pragma unroll
    for (int s = 0; s < 5; ++s) {
      int t  = wave + 8 * s;
      int ri = t >> 2, cj = t & 3;
#pragma unroll
      for (int v = 0; v < 8; ++v) {
        int n = ri * 16 + v + 8 * hi;
        int k = cj * 16 + lr;
        if (n < NPTS && k < NCLU && ohT[k * OPITCH + n] > 0.f) {
          unsigned long long key =
              (((unsigned long long)__float_as_uint(ic[s][v])) << 32) |
              (unsigned int)n;
          atomicMin(&keys[k], key);
        }
      }
    }
    __syncthreads();
    if (tid < NCLU) med[tid] = (int)(unsigned int)(keys[tid] & 0xffffffffull);
    __syncthreads();
  }

  // ================= Phase E: sort medoids ascending =================
  if (tid == 0) {
    for (int i = 1; i < NCLU; ++i) {
      int key = med[i], j = i - 1;
      while (j >= 0 && med[j] > key) { med[j + 1] = med[j]; j--; }
      med[j + 1] = key;
    }
  }
  __syncthreads();

  // ================= Phase F: outputs (float4 streams) =================
  const int col = bi * 4 + ta;                 // batch-major output column
  {                                            // CLS: mean over the 3 frames
    const float4* c0 = (const float4*)(x + (long)frame0 * WIDTH);
    float4* dst = (float4*)(out + (long)col * WIDTH);
    for (int d = tid; d < WIDTH / 4; d += 256) {
      float4 a = c0[d], b = c0[WIDTH / 4 + d], c = c0[2 * (WIDTH / 4) + d];
      float4 v;
      v.x = (a.x + b.x + c.x) / 3.0f;
      v.y = (a.y + b.y + c.y) / 3.0f;
      v.z = (a.z + b.z + c.z) / 3.0f;
      v.w = (a.w + b.w + c.w) / 3.0f;
      dst[d] = v;
    }
  }
#pragma unroll 1
  for (int k = 0; k < NCLU; ++k) {             // medoid token rows
    const float4* src = (const float4*)(x + rowoff[med[k]]);
    float4* dst = (float4*)(out + ((long)(k + 1) * 256 + col) * WIDTH);
    for (int d = tid; d < WIDTH / 4; d += 256) dst[d] = src[d];
  }
}

extern "C" void kernel_launch(void* const* d_in, const int* in_sizes, int n_in,
                              void* d_out, int out_size, void* d_ws, size_t ws_size,
                              hipStream_t stream) {
  (void)in_sizes; (void)n_in; (void)d_ws; (void)ws_size; (void)out_size;
  const float* x = (const float*)d_in[0];     // [50, 768, 768] fp32
  float* out = (float*)d_out;                 // [50, 256, 768] fp32
  const size_t smem = (size_t)LDS_FLOATS * sizeof(float);  // 184,480 B
  (void)hipFuncSetAttribute((const void*)tci_kernel,
                            hipFuncAttributeMaxDynamicSharedMemorySize,
                            (int)smem);
  tci_kernel<<<256, 256, smem, stream>>>(x, out);
}